// RandomMaskSubgraphs_65575560675337
// MI455X (gfx1250) — compile-verified
//
#include <hip/hip_runtime.h>
#include <hip/hip_bf16.h>
#include <stdint.h>

// ---------------------------------------------------------------------------
// Problem constants (from the reference)
// ---------------------------------------------------------------------------
#define N_NODES   8192
#define NNZ_E     262144
#define NSEEDS    256
#define SAMP_NUM  7372                      // int(8192 * 0.9)
#define BM_WPR    (N_NODES / 32)            // bitmap words per row = 256
#define KSEG      8                         // K-split for the degree pass
#define KSEG_COLS (N_NODES / KSEG)          // 1024 columns per segment
#define TILE_K    64                        // columns staged per LDS tile

typedef float v2f __attribute__((ext_vector_type(2)));
typedef float v8f __attribute__((ext_vector_type(8)));

#if __has_builtin(__builtin_amdgcn_wmma_f32_16x16x4_f32)
#define HAVE_WMMA_F32_16X16X4 1
#else
#define HAVE_WMMA_F32_16X16X4 0
#endif

// ---------------------------------------------------------------------------
// threefry2x32-20 (JAX PRNG core), plus JAX key(42) split-3 / bit derivation
// ---------------------------------------------------------------------------
__device__ __forceinline__ void tf_round(uint32_t& x0, uint32_t& x1, int r) {
  x0 += x1;
  x1 = (x1 << r) | (x1 >> (32 - r));
  x1 ^= x0;
}

__device__ inline uint2 threefry2x32(uint32_t k0, uint32_t k1,
                                     uint32_t c0, uint32_t c1) {
  const uint32_t ks2 = k0 ^ k1 ^ 0x1BD11BDAu;
  uint32_t x0 = c0 + k0, x1 = c1 + k1;
  tf_round(x0, x1, 13); tf_round(x0, x1, 15); tf_round(x0, x1, 26); tf_round(x0, x1, 6);
  x0 += k1;  x1 += ks2 + 1u;
  tf_round(x0, x1, 17); tf_round(x0, x1, 29); tf_round(x0, x1, 16); tf_round(x0, x1, 24);
  x0 += ks2; x1 += k0 + 2u;
  tf_round(x0, x1, 13); tf_round(x0, x1, 15); tf_round(x0, x1, 26); tf_round(x0, x1, 6);
  x0 += k0;  x1 += k1 + 3u;
  tf_round(x0, x1, 17); tf_round(x0, x1, 29); tf_round(x0, x1, 16); tf_round(x0, x1, 24);
  x0 += k1;  x1 += ks2 + 4u;
  tf_round(x0, x1, 13); tf_round(x0, x1, 15); tf_round(x0, x1, 26); tf_round(x0, x1, 6);
  x0 += ks2; x1 += k0 + 5u;
  uint2 r; r.x = x0; r.y = x1;
  return r;
}

// jax.random.split(key(42), 3): bits = threefry(key, iota(6) split in halves)
__device__ inline void jax_key42_split3(uint2& ka, uint2& kb, uint2& kc) {
  uint2 t0 = threefry2x32(0u, 42u, 0u, 3u);
  uint2 t1 = threefry2x32(0u, 42u, 1u, 4u);
  uint2 t2 = threefry2x32(0u, 42u, 2u, 5u);
  ka.x = t0.x; ka.y = t1.x;     // bits[0], bits[1]
  kb.x = t2.x; kb.y = t0.y;     // bits[2], bits[3]
  kc.x = t1.y; kc.y = t2.y;     // bits[4], bits[5]
}

// i-th word of random_bits(key, (n,)) for even n (counts split into halves)
__device__ inline uint32_t jax_bits(uint2 key, uint32_t i, uint32_t n) {
  const uint32_t half = n >> 1;
  if (i < half) return threefry2x32(key.x, key.y, i, half + i).x;
  return threefry2x32(key.x, key.y, i - half, i).y;
}

__device__ inline float bits_to_unit(uint32_t b) {   // uniform in [0,1)
  return __uint_as_float((b >> 9) | 0x3f800000u) - 1.0f;
}

// ---------------------------------------------------------------------------
// Small utility / graph kernels
// ---------------------------------------------------------------------------
__global__ void k_fill_i32(int* __restrict__ p, int v, size_t n) {
  size_t i = (size_t)blockIdx.x * blockDim.x + threadIdx.x;
  const size_t st = (size_t)gridDim.x * blockDim.x;
  for (; i < n; i += st) p[i] = v;
}

__global__ void k_seed(const int* __restrict__ seeds,
                       int* __restrict__ seed_mask, int* __restrict__ mask_nodes) {
  const int t = threadIdx.x;
  if (t < NSEEDS) {
    const int s = seeds[t];
    seed_mask[s] = 1;
    mask_nodes[s] = 1;
  }
}

// one BFS-edge-removal sweep; optionally marks touched nodes in `nxt`
__global__ void k_bfs(const int* __restrict__ rows, const int* __restrict__ cols,
                      int* __restrict__ keep, const int* __restrict__ sm,
                      int* __restrict__ nxt, int markNodes) {
  const int e = blockIdx.x * blockDim.x + threadIdx.x;
  if (e >= NNZ_E) return;
  const int r = rows[e], c = cols[e];
  if (keep[e] && (sm[r] | sm[c])) {
    keep[e] = 0;
    if (markNodes) { nxt[r] = 1; nxt[c] = 1; }   // idempotent stores
  }
}

__global__ void k_nodes(const int* __restrict__ nxt,
                        int* __restrict__ seed_mask, int* __restrict__ mask_nodes) {
  const int i = blockIdx.x * blockDim.x + threadIdx.x;
  if (i < N_NODES) {
    const int v = nxt[i] ? 1 : 0;
    seed_mask[i] = v;             // seed_mask := nxt (replaced, per reference)
    if (v) mask_nodes[i] = 1;
  }
}

__global__ void k_sample(int* __restrict__ mask_nodes) {
  const int t = blockIdx.x * blockDim.x + threadIdx.x;
  if (t >= SAMP_NUM) return;
  uint2 ka, kb, kc;
  jax_key42_split3(ka, kb, kc);
  const uint32_t b = jax_bits(ka, (uint32_t)t, SAMP_NUM);  // randint(0,8192): 8192=2^13
  mask_nodes[b & (N_NODES - 1)] = 1;
}

// compact ascending indices of mask_nodes into mask_idx (single 1024-thread block)
__global__ void __launch_bounds__(1024) k_scan(const int* __restrict__ mn,
                                               int* __restrict__ mask_idx,
                                               int* __restrict__ tem_num) {
  __shared__ int cnt[1024];
  const int t = threadIdx.x;
  const int base = t * 8;
  int local[8]; int c = 0;
  #pragma unroll
  for (int k = 0; k < 8; ++k) { local[k] = mn[base + k]; c += (local[k] != 0); }
  cnt[t] = c;
  __syncthreads();
  for (int off = 1; off < 1024; off <<= 1) {
    const int v = (t >= off) ? cnt[t - off] : 0;
    __syncthreads();
    cnt[t] += v;
    __syncthreads();
  }
  int start = cnt[t] - c;
  #pragma unroll
  for (int k = 0; k < 8; ++k) if (local[k]) mask_idx[start++] = base + k;
  const int total = cnt[1023];
  if (t == 1023) *tem_num = total;
  for (int i = total + t; i < N_NODES; i += 1024) mask_idx[i] = 0;  // fill_value=0
}

// ---------------------------------------------------------------------------
// Coverage bitmaps (1 bit per (i,j)); atomicOr is idempotent -> deterministic
// ---------------------------------------------------------------------------
__global__ void k_cov_edges(const int* __restrict__ rows, const int* __restrict__ cols,
                            const int* __restrict__ keep,
                            uint32_t* __restrict__ enc_bm, uint32_t* __restrict__ dec_bm) {
  const int e = blockIdx.x * blockDim.x + threadIdx.x;
  if (e >= NNZ_E || !keep[e]) return;
  const int r = rows[e], c = cols[e];
  const uint32_t bit = 1u << (c & 31);
  const size_t wi = (size_t)r * BM_WPR + (c >> 5);
  atomicOr(&enc_bm[wi], bit);
  atomicOr(&dec_bm[wi], bit);
}

__global__ void k_cov_self(uint32_t* __restrict__ dec_bm) {
  const int i = blockIdx.x * blockDim.x + threadIdx.x;
  if (i < N_NODES)
    atomicOr(&dec_bm[(size_t)i * BM_WPR + (i >> 5)], 1u << (i & 31));
}

__global__ void k_cov_rand(const int* __restrict__ mask_idx,
                           const int* __restrict__ tem_num_p,
                           uint32_t* __restrict__ dec_bm) {
  const int e = blockIdx.x * blockDim.x + threadIdx.x;
  if (e >= NNZ_E) return;
  uint2 ka, kb, kc;
  jax_key42_split3(ka, kb, kc);
  const int tem = *tem_num_p;
  const float ft = (float)tem;
  const float u1 = bits_to_unit(jax_bits(kb, (uint32_t)e, NNZ_E));
  const float u2 = bits_to_unit(jax_bits(kc, (uint32_t)e, NNZ_E));
  int ti = (int)(u1 * ft); ti = ti < tem - 1 ? ti : tem - 1;
  int tj = (int)(u2 * ft); tj = tj < tem - 1 ? tj : tem - 1;
  const int tr = mask_idx[ti], tc = mask_idx[tj];
  atomicOr(&dec_bm[(size_t)tr * BM_WPR + (tc >> 5)], 1u << (tc & 31));
  atomicOr(&dec_bm[(size_t)tc * BM_WPR + (tr >> 5)], 1u << (tr & 31));
}

// ---------------------------------------------------------------------------
// Degree pass: rowsum(enc_cov ? comp : 0) expressed as (16x4 tile) x ones via
// V_WMMA_F32_16X16X4_F32, with tiles staged global->LDS through the CDNA5
// async path (global_load_async_to_lds_b128 + s_wait_asynccnt).
// One wave per block; grid = (N/16 row tiles) x (KSEG column segments).
// Partials stored per-segment and reduced in fixed order -> deterministic.
// ---------------------------------------------------------------------------
__global__ void __launch_bounds__(32) k_deg(const float* __restrict__ comp,
                                            const uint32_t* __restrict__ enc_bm,
                                            float* __restrict__ deg_part) {
  __shared__ float tile[16 * TILE_K];                 // 4 KB staging tile
  const int lane    = threadIdx.x;                    // wave32
  const int rowBase = blockIdx.x * 16;
  const int kseg    = blockIdx.y;
  const int M  = lane & 15;                           // A-fragment row
  const int Kh = lane >> 4;                           // A-fragment K-half
  const uint32_t ldsBase = (uint32_t)(uintptr_t)&tile[0];

#if HAVE_WMMA_F32_16X16X4
  v8f acc = {0.f, 0.f, 0.f, 0.f, 0.f, 0.f, 0.f, 0.f};
  v2f bOnes; bOnes.x = 1.0f; bOnes.y = 1.0f;          // B = ones(4x16)
#else
  float degacc = 0.0f;
#endif

  const int kBeg = kseg * KSEG_COLS;
  const int kEnd = kBeg + KSEG_COLS;
  for (int k0 = kBeg; k0 < kEnd; k0 += TILE_K) {
    // stage 16 rows x 64 f32 (4 KB): 8 async issues x 32 lanes x 16 B, coalesced
    #pragma unroll
    for (int j = 0; j < 8; ++j) {
      const int e = j * 32 + lane;                    // 256 sixteen-byte segments
      const int r = e >> 4;                           // 16 segments per row
      const int s = e & 15;
      const char* g = (const char*)comp +
          ((size_t)(rowBase + r) * N_NODES + (size_t)k0) * 4u + (size_t)s * 16u;
      const uint32_t l = ldsBase + (uint32_t)(r * TILE_K * 4 + s * 16);
      asm volatile("global_load_async_to_lds_b128 %0, %1, off"
                   :: "v"(l), "v"(g) : "memory");
    }
    asm volatile("s_wait_asynccnt 0x0" ::: "memory");

    const size_t bmRow = (size_t)(rowBase + M) * BM_WPR;
    const uint32_t w0 = enc_bm[bmRow + (k0 >> 5)];
    const uint32_t w1 = enc_bm[bmRow + (k0 >> 5) + 1];

#if HAVE_WMMA_F32_16X16X4
    #pragma unroll
    for (int t = 0; t < 16; ++t) {                    // 16 WMMAs of K=4
      const int col = t * 4 + Kh * 2;
      const uint32_t w = (t < 8) ? w0 : w1;
      const int bp = col & 31;
      v2f a;
      a.x = ((w >> bp)       & 1u) ? tile[M * TILE_K + col]     : 0.0f;
      a.y = ((w >> (bp + 1)) & 1u) ? tile[M * TILE_K + col + 1] : 0.0f;
      acc = __builtin_amdgcn_wmma_f32_16x16x4_f32(false, a, false, bOnes,
                                                  (short)0, acc, false, false);
    }
#else
    if (Kh == 0) {
      for (int c = 0; c < TILE_K; ++c) {
        const uint32_t w = (c < 32) ? w0 : w1;
        if ((w >> (c & 31)) & 1u) degacc += tile[M * TILE_K + c];
      }
    }
#endif
  }

#if HAVE_WMMA_F32_16X16X4
  // C/D layout: lanes 0-15 VGPR v holds row M=v; lanes 16-31 hold M=v+8;
  // columns are replicated row-sums, so lanes with N==0 (M==0) extract them.
  if (M == 0) {
    #pragma unroll
    for (int v = 0; v < 8; ++v)
      deg_part[(size_t)kseg * N_NODES + rowBase + Kh * 8 + v] = acc[v];
  }
#else
  if (Kh == 0) deg_part[(size_t)kseg * N_NODES + rowBase + M] = degacc;
#endif
}

__global__ void k_norm(const float* __restrict__ deg_part, float* __restrict__ norm) {
  const int i = blockIdx.x * blockDim.x + threadIdx.x;
  if (i < N_NODES) {
    float d = 0.0f;
    #pragma unroll
    for (int s = 0; s < KSEG; ++s) d += deg_part[(size_t)s * N_NODES + i]; // fixed order
    norm[i] = 1.0f / sqrtf(d + 1e-12f);
  }
}

// ---------------------------------------------------------------------------
// Final bandwidth pass: one float4 per thread-iteration; 4 coverage bits per
// bitmap word lookup. ~784 MB moved -> ~34 us roofline at 23.3 TB/s.
// ---------------------------------------------------------------------------
__global__ void k_final(const float* __restrict__ comp,
                        const uint32_t* __restrict__ enc_bm,
                        const uint32_t* __restrict__ dec_bm,
                        const float* __restrict__ norm,
                        float* __restrict__ enc_out, float* __restrict__ dec_out) {
  const size_t nq = (size_t)N_NODES * (N_NODES / 4);
  const size_t stride = (size_t)gridDim.x * blockDim.x;
  for (size_t q = (size_t)blockIdx.x * blockDim.x + threadIdx.x; q < nq; q += stride) {
    const int i  = (int)(q >> 11);                    // N/4 == 2048 == 2^11
    const int j4 = (int)(q & 2047);
    const int jb = j4 << 2;
    const float4 c  = ((const float4*)comp)[q];
    const float4 nj = ((const float4*)norm)[j4];
    const float  ni = norm[i];
    const uint32_t we = enc_bm[(size_t)i * BM_WPR + (jb >> 5)];
    const uint32_t wd = dec_bm[(size_t)i * BM_WPR + (jb >> 5)];
    const int bp = jb & 31;
    float4 e, d;
    e.x = ((we >> (bp + 0)) & 1u) ? c.x * ni * nj.x : 0.0f;
    e.y = ((we >> (bp + 1)) & 1u) ? c.y * ni * nj.y : 0.0f;
    e.z = ((we >> (bp + 2)) & 1u) ? c.z * ni * nj.z : 0.0f;
    e.w = ((we >> (bp + 3)) & 1u) ? c.w * ni * nj.w : 0.0f;
    d.x = ((wd >> (bp + 0)) & 1u) ? c.x : 0.0f;
    d.y = ((wd >> (bp + 1)) & 1u) ? c.y : 0.0f;
    d.z = ((wd >> (bp + 2)) & 1u) ? c.z : 0.0f;
    d.w = ((wd >> (bp + 3)) & 1u) ? c.w : 0.0f;
    ((float4*)enc_out)[q] = e;
    ((float4*)dec_out)[q] = d;
  }
}

// ---------------------------------------------------------------------------
// Host-side orchestration (graph-capture safe: kernels only, same every call)
// ---------------------------------------------------------------------------
extern "C" void kernel_launch(void* const* d_in, const int* in_sizes, int n_in,
                              void* d_out, int out_size, void* d_ws, size_t ws_size,
                              hipStream_t stream) {
  (void)in_sizes; (void)n_in; (void)out_size; (void)ws_size;
  const int*   adj_rows = (const int*)d_in[0];
  const int*   adj_cols = (const int*)d_in[1];
  // d_in[2] = adj_values: unused (reference gathers values from complemental)
  const int*   seeds    = (const int*)d_in[3];
  const float* comp     = (const float*)d_in[4];
  float* enc_out = (float*)d_out;
  float* dec_out = enc_out + (size_t)N_NODES * N_NODES;

  // workspace carve-up (256 B aligned chunks), ~18 MB total
  char* w = (char*)d_ws;
  size_t off = 0;
  auto take = [&](size_t bytes) -> char* {
    char* p = w + off;
    off = (off + bytes + 255) & ~(size_t)255;
    return p;
  };
  int*      keep       = (int*)take((size_t)NNZ_E * 4);
  int*      seed_mask  = (int*)take((size_t)N_NODES * 4);
  int*      nxt        = (int*)take((size_t)N_NODES * 4);
  int*      mask_nodes = (int*)take((size_t)N_NODES * 4);
  int*      mask_idx   = (int*)take((size_t)N_NODES * 4);
  int*      tem_num    = (int*)take(256);
  uint32_t* enc_bm     = (uint32_t*)take((size_t)N_NODES * BM_WPR * 4);
  uint32_t* dec_bm     = (uint32_t*)take((size_t)N_NODES * BM_WPR * 4);
  float*    deg_part   = (float*)take((size_t)KSEG * N_NODES * 4);
  float*    norm       = (float*)take((size_t)N_NODES * 4);

  // init
  k_fill_i32<<<1024, 256, 0, stream>>>(keep, 1, (size_t)NNZ_E);
  k_fill_i32<<<32, 256, 0, stream>>>(seed_mask, 0, (size_t)N_NODES);
  k_fill_i32<<<32, 256, 0, stream>>>(nxt, 0, (size_t)N_NODES);
  k_fill_i32<<<32, 256, 0, stream>>>(mask_nodes, 0, (size_t)N_NODES);
  k_fill_i32<<<2048, 256, 0, stream>>>((int*)enc_bm, 0, (size_t)N_NODES * BM_WPR);
  k_fill_i32<<<2048, 256, 0, stream>>>((int*)dec_bm, 0, (size_t)N_NODES * BM_WPR);

  // BFS edge masking (MASK_DEPTH == 2)
  k_seed<<<1, 256, 0, stream>>>(seeds, seed_mask, mask_nodes);
  k_bfs<<<NNZ_E / 256, 256, 0, stream>>>(adj_rows, adj_cols, keep, seed_mask, nxt, 1);
  k_nodes<<<N_NODES / 256, 256, 0, stream>>>(nxt, seed_mask, mask_nodes);
  k_bfs<<<NNZ_E / 256, 256, 0, stream>>>(adj_rows, adj_cols, keep, seed_mask, nxt, 0);

  // random node sampling + index compaction
  k_sample<<<(SAMP_NUM + 255) / 256, 256, 0, stream>>>(mask_nodes);
  k_scan<<<1, 1024, 0, stream>>>(mask_nodes, mask_idx, tem_num);

  // coverage bitmaps
  k_cov_edges<<<NNZ_E / 256, 256, 0, stream>>>(adj_rows, adj_cols, keep, enc_bm, dec_bm);
  k_cov_self<<<N_NODES / 256, 256, 0, stream>>>(dec_bm);
  k_cov_rand<<<NNZ_E / 256, 256, 0, stream>>>(mask_idx, tem_num, dec_bm);

  // degree / norm (WMMA + async global->LDS staging)
  dim3 gdeg(N_NODES / 16, KSEG);
  k_deg<<<gdeg, 32, 0, stream>>>(comp, enc_bm, deg_part);
  k_norm<<<N_NODES / 256, 256, 0, stream>>>(deg_part, norm);

  // final bandwidth-bound masked writes
  k_final<<<8192, 256, 0, stream>>>(comp, enc_bm, dec_bm, norm, enc_out, dec_out);
}